// GNNModel_3307124818507
// MI455X (gfx1250) — compile-verified
//
#include <hip/hip_runtime.h>
#include <hip/hip_bf16.h>
#include <stdint.h>

// GCN: 3 x [ GEMM (WMMA f32 16x16x4) -> edge scatter-add -> selfloop+bias+relu ]
// N=100000, DIN=H=128, E=1600000 (derived from in_sizes at runtime).

typedef __attribute__((ext_vector_type(2))) float v2f;
typedef __attribute__((ext_vector_type(8))) float v8f;

#define HDIM 128

// ---------------- degree / dinv ----------------
__global__ void deg_init_kernel(float* __restrict__ deg, int n) {
    int i = blockIdx.x * blockDim.x + threadIdx.x;
    if (i < n) deg[i] = 1.0f;  // self-loop weight
}

__global__ void deg_acc_kernel(const long long* __restrict__ col,
                               const float* __restrict__ ew,
                               float* __restrict__ deg, long long ne) {
    long long e = (long long)blockIdx.x * blockDim.x + threadIdx.x;
    if (e < ne) {
        atomicAdd(&deg[(int)col[e]], ew[e]);
    }
}

__global__ void deg_rsqrt_kernel(float* __restrict__ deg, int n) {
    int i = blockIdx.x * blockDim.x + threadIdx.x;
    if (i < n) deg[i] = rsqrtf(deg[i]);  // deg >= 1, safe
}

// ---------------- GEMM: Y[nrows x 128] = X[nrows x 128] @ W[128 x 128] ------
// One 16x16 output tile per wave via V_WMMA_F32_16X16X4_F32 (32 K-steps).
// Block = 256 threads = 8 waves; wave w owns column tile w (8*16 = 128 cols).
// Grid.x = nrows/16 row tiles. nrows divisible by 16 (100000 = 6250*16),
// so no divergence anywhere -> EXEC all-1s as WMMA requires.
__global__ __launch_bounds__(256) void gemm128_wmma_kernel(
    const float* __restrict__ X, const float* __restrict__ W,
    float* __restrict__ Y) {
    const int lane = threadIdx.x & 31;
    const int wave = threadIdx.x >> 5;
    const int lm   = lane & 15;   // M (for A/C) or N (for B/C) index
    const int hi   = lane >> 4;   // lane half selector

    const int m0 = blockIdx.x * 16;
    const int n0 = wave * 16;

    const float* __restrict__ xrow = X + (size_t)(m0 + lm) * HDIM;

    v8f c = {};
    #pragma unroll 4
    for (int k4 = 0; k4 < HDIM; k4 += 4) {
        // A 16x4 fp32: lanes 0-15 -> {K=k4, k4+1}, lanes 16-31 -> {k4+2, k4+3}
        v2f a;
        a.x = xrow[k4 + 2 * hi + 0];
        a.y = xrow[k4 + 2 * hi + 1];
        // B 4x16 fp32: VGPR j holds row K = k4 + j + 2*hi, N = lm
        v2f b;
        b.x = W[(size_t)(k4 + 2 * hi + 0) * HDIM + n0 + lm];
        b.y = W[(size_t)(k4 + 2 * hi + 1) * HDIM + n0 + lm];
        c = __builtin_amdgcn_wmma_f32_16x16x4_f32(
            /*neg_a=*/false, a, /*neg_b=*/false, b,
            /*c_mod=*/(short)0, c, /*reuse_a=*/false, /*reuse_b=*/false);
    }

    // C/D 16x16 fp32: VGPR j -> row M = j + 8*hi, col N = lm
    #pragma unroll
    for (int j = 0; j < 8; ++j) {
        Y[(size_t)(m0 + j + 8 * hi) * HDIM + n0 + lm] = c[j];
    }
}

// ---------------- zero fill ----------------
__global__ void zero_kernel(float* __restrict__ p, size_t n) {
    size_t i = (size_t)blockIdx.x * blockDim.x + threadIdx.x;
    if (i < n) p[i] = 0.0f;
}

// ---------------- edge scatter: agg[col] += dinv[row]*ew*dinv[col] * xw[row]
// One wave per edge: 32 lanes x float4 = 128 features. Gathers are
// 128B-coalesced per wave; scatter uses global_atomic_add_f32 (L2-resident).
__global__ __launch_bounds__(256) void edge_agg_kernel(
    const long long* __restrict__ row, const long long* __restrict__ col,
    const float* __restrict__ ew, const float* __restrict__ dinv,
    const float* __restrict__ xw, float* __restrict__ agg, long long ne) {
    long long e = (long long)blockIdx.x * 8 + (threadIdx.x >> 5);
    if (e >= ne) return;
    const int lane = threadIdx.x & 31;
    const int r = (int)row[e];
    const int c = (int)col[e];
    const float norm = dinv[r] * ew[e] * dinv[c];
    const float4 v = ((const float4*)(xw + (size_t)r * HDIM))[lane];
    float* dst = agg + (size_t)c * HDIM + lane * 4;
    atomicAdd(dst + 0, norm * v.x);
    atomicAdd(dst + 1, norm * v.y);
    atomicAdd(dst + 2, norm * v.z);
    atomicAdd(dst + 3, norm * v.w);
}

// ---------------- out = relu(agg + dinv^2 * xw + bias) ----------------
__global__ __launch_bounds__(256) void finalize_kernel(
    const float* __restrict__ agg, const float* __restrict__ xw,
    const float* __restrict__ dinv, const float* __restrict__ bias,
    float* __restrict__ out, size_t total) {
    size_t idx = (size_t)blockIdx.x * blockDim.x + threadIdx.x;
    if (idx >= total) return;
    const int h = (int)(idx & (HDIM - 1));
    const int i = (int)(idx >> 7);
    const float d  = dinv[i];
    const float v  = agg[idx] + (d * d) * xw[idx] + bias[h];
    out[idx] = fmaxf(v, 0.0f);
}

// ---------------- driver ----------------
extern "C" void kernel_launch(void* const* d_in, const int* in_sizes, int n_in,
                              void* d_out, int out_size, void* d_ws, size_t ws_size,
                              hipStream_t stream) {
    const float*     x     = (const float*)d_in[0];
    const long long* eidx  = (const long long*)d_in[1];  // [2, E] int64
    const float*     eattr = (const float*)d_in[2];
    const float*     W1    = (const float*)d_in[3];
    const float*     b1    = (const float*)d_in[4];
    const float*     W2    = (const float*)d_in[5];
    const float*     b2    = (const float*)d_in[6];
    const float*     W3    = (const float*)d_in[7];
    const float*     b3    = (const float*)d_in[8];
    float*           out   = (float*)d_out;

    const long long E_ = (long long)in_sizes[2];
    const int       N_ = in_sizes[0] / HDIM;
    const size_t    NH = (size_t)N_ * HDIM;

    const long long* row = eidx;       // edge_index[0]
    const long long* col = eidx + E_;  // edge_index[1]

    // Workspace layout: dinv | xw | agg | hA   (~154 MB for N=100k)
    float* dinv = (float*)d_ws;
    float* xw   = dinv + (((size_t)N_ + 255) & ~(size_t)255);
    float* agg  = xw + NH;
    float* hA   = agg + NH;

    const int TB = 256;

    // deg -> dinv (recomputed every call; deterministic)
    deg_init_kernel<<<(N_ + TB - 1) / TB, TB, 0, stream>>>(dinv, N_);
    deg_acc_kernel<<<(unsigned)((E_ + TB - 1) / TB), TB, 0, stream>>>(col, eattr, dinv, E_);
    deg_rsqrt_kernel<<<(N_ + TB - 1) / TB, TB, 0, stream>>>(dinv, N_);

    const unsigned gemm_grid = (unsigned)(N_ / 16);            // 6250 (exact)
    const unsigned edge_grid = (unsigned)((E_ + 7) / 8);       // 8 edges/block
    const unsigned elem_grid = (unsigned)((NH + TB - 1) / TB); // pointwise

    const float* layer_in = x;
    const float* Ws[3] = {W1, W2, W3};
    const float* bs[3] = {b1, b2, b3};

    for (int L = 0; L < 3; ++L) {
        float* layer_out = (L == 2) ? out : hA;
        gemm128_wmma_kernel<<<gemm_grid, TB, 0, stream>>>(layer_in, Ws[L], xw);
        zero_kernel<<<elem_grid, TB, 0, stream>>>(agg, NH);
        edge_agg_kernel<<<edge_grid, TB, 0, stream>>>(row, col, eattr, dinv, xw, agg, E_);
        finalize_kernel<<<elem_grid, TB, 0, stream>>>(agg, xw, dinv, bs[L], layer_out, NH);
        layer_in = layer_out;  // hA consumed by next layer's GEMM before overwrite
    }
}